// TrendAwareLoss_49254684950871
// MI455X (gfx1250) — compile-verified
//
#include <hip/hip_runtime.h>
#include <hip/hip_bf16.h>
#include <stdint.h>

typedef __attribute__((ext_vector_type(2))) float v2f;
typedef __attribute__((ext_vector_type(8))) float v8f;

#define WAVES_PER_BLOCK   8
#define BLOCK_THREADS     256
#define MAX_PARTIAL_BLOCKS 4096

__device__ __forceinline__ float wave_max32(float v) {
#pragma unroll
    for (int off = 16; off > 0; off >>= 1)
        v = fmaxf(v, __shfl_xor(v, off, 32));
    return v;
}

// Exact f32 cross-lane (32-lane) sum. Uses V_WMMA_F32_16X16X4_F32 with B = ones:
// A[m,0]=partial[lane m], A[m,2]=partial[lane m+16], A[m,1]=A[m,3]=0
// => D[m,n] = partial[m] + partial[m+16] for every column n.
// Lane L (<16) holds D[0..7, L] in the 8 accumulator VGPRs, lane L+16 holds D[8..15, L].
// Summing the 8 VGPRs + one xor-16 shuffle add yields the full 32-lane sum on all lanes.
__device__ __forceinline__ float wave_sum32(float partial) {
#if defined(__gfx1250__) && __has_builtin(__builtin_amdgcn_wmma_f32_16x16x4_f32)
    v2f a; a.x = partial; a.y = 0.0f;
    v2f b; b.x = 1.0f;    b.y = 1.0f;
    v8f c = {};
    v8f d = __builtin_amdgcn_wmma_f32_16x16x4_f32(false, a, false, b, (short)0, c,
                                                  false, false);
    float s = 0.0f;
#pragma unroll
    for (int i = 0; i < 8; ++i) s += d[i];
    s += __shfl_xor(s, 16, 32);
    return s;
#else
#pragma unroll
    for (int off = 16; off > 0; off >>= 1)
        partial += __shfl_xor(partial, off, 32);
    return partial;
#endif
}

__global__ __launch_bounds__(BLOCK_THREADS)
void ce_partial_kernel(const float* __restrict__ logits,
                       const long long* __restrict__ labels,
                       float* __restrict__ partials,  // [gridDim.x * 2]
                       int B, int C) {
    __shared__ float s_num[WAVES_PER_BLOCK];
    __shared__ float s_den[WAVES_PER_BLOCK];

    const int lane  = threadIdx.x & 31;
    const int wave  = threadIdx.x >> 5;
    const int wgid  = blockIdx.x * WAVES_PER_BLOCK + wave;  // global wave id
    const int wstep = gridDim.x * WAVES_PER_BLOCK;

    float acc_num = 0.0f;
    float acc_den = 0.0f;

    const int c4 = C >> 2;  // float4s per row

    for (int row = wgid; row < B; row += wstep) {
        float m, s;
        if (C == 1024) {
            // Fast path: whole row (4 KB) in registers, one memory pass.
            const float4* rp =
                reinterpret_cast<const float4*>(logits) + (size_t)row * c4;
            float4 vals[8];
            m = -INFINITY;
#pragma unroll
            for (int ch = 0; ch < 8; ++ch) {
                vals[ch] = rp[ch * 32 + lane];  // coalesced 512B per chunk
                m = fmaxf(m, fmaxf(fmaxf(vals[ch].x, vals[ch].y),
                                   fmaxf(vals[ch].z, vals[ch].w)));
            }
            m = wave_max32(m);
            float p = 0.0f;
#pragma unroll
            for (int ch = 0; ch < 8; ++ch) {
                p += __expf(vals[ch].x - m);
                p += __expf(vals[ch].y - m);
                p += __expf(vals[ch].z - m);
                p += __expf(vals[ch].w - m);
            }
            s = wave_sum32(p);
        } else {
            // Generic fallback: two passes (second hits cache).
            const float* rp = logits + (size_t)row * C;
            m = -INFINITY;
            for (int i = lane; i < C; i += 32) m = fmaxf(m, rp[i]);
            m = wave_max32(m);
            float p = 0.0f;
            for (int i = lane; i < C; i += 32) p += __expf(rp[i] - m);
            s = wave_sum32(p);
        }

        if (lane == 0) {
            long long lbl = labels[row];
            // weight = upper_bound(labels, lbl) - row   (labels sorted ascending)
            int lo = row + 1, hi = B;
            while (lo < hi) {
                int mid = (lo + hi) >> 1;
                if (labels[mid] <= lbl) lo = mid + 1; else hi = mid;
            }
            float w = (float)(lo - row);
            // JAX take_along_axis clamps out-of-bounds indices
            int ci = (int)lbl; if (ci > C - 1) ci = C - 1; if (ci < 0) ci = 0;
            float xl = logits[(size_t)row * C + ci];
            float ce = m + __logf(s) - xl;
            acc_num += ce * w;
            acc_den += w;
        }
    }

    if (lane == 0) { s_num[wave] = acc_num; s_den[wave] = acc_den; }
    __syncthreads();
    if (threadIdx.x == 0) {
        float n = 0.0f, d = 0.0f;
#pragma unroll
        for (int i = 0; i < WAVES_PER_BLOCK; ++i) { n += s_num[i]; d += s_den[i]; }
        partials[2 * blockIdx.x + 0] = n;
        partials[2 * blockIdx.x + 1] = d;
    }
}

__global__ __launch_bounds__(BLOCK_THREADS)
void finalize_kernel(const float* __restrict__ partials, int nblocks,
                     float* __restrict__ out) {
    __shared__ float sn[BLOCK_THREADS];
    __shared__ float sd[BLOCK_THREADS];
    float n = 0.0f, d = 0.0f;
    for (int i = threadIdx.x; i < nblocks; i += BLOCK_THREADS) {
        n += partials[2 * i + 0];
        d += partials[2 * i + 1];
    }
    sn[threadIdx.x] = n; sd[threadIdx.x] = d;
    __syncthreads();
#pragma unroll
    for (int off = BLOCK_THREADS / 2; off > 0; off >>= 1) {
        if (threadIdx.x < off) {
            sn[threadIdx.x] += sn[threadIdx.x + off];
            sd[threadIdx.x] += sd[threadIdx.x + off];
        }
        __syncthreads();
    }
    if (threadIdx.x == 0) out[0] = sn[0] / (sd[0] + 1e-12f);
}

extern "C" void kernel_launch(void* const* d_in, const int* in_sizes, int n_in,
                              void* d_out, int out_size, void* d_ws, size_t ws_size,
                              hipStream_t stream) {
    const float*     logits = (const float*)d_in[0];
    const long long* labels = (const long long*)d_in[1];
    const int B = in_sizes[1];
    const int C = in_sizes[0] / B;

    float* partials = (float*)d_ws;

    int nblocks = (B + WAVES_PER_BLOCK - 1) / WAVES_PER_BLOCK;
    if (nblocks > MAX_PARTIAL_BLOCKS) nblocks = MAX_PARTIAL_BLOCKS;
    if (nblocks < 1) nblocks = 1;

    ce_partial_kernel<<<nblocks, BLOCK_THREADS, 0, stream>>>(
        logits, labels, partials, B, C);
    finalize_kernel<<<1, BLOCK_THREADS, 0, stream>>>(
        partials, nblocks, (float*)d_out);
}